// GCN_62362925138630
// MI455X (gfx1250) — compile-verified
//
#include <hip/hip_runtime.h>
#include <math.h>

// ---------------------------------------------------------------------------
// GCN 2-layer forward, 5 graphs. MI455X (gfx1250, wave32).
//
// Design: memory-bound problem (~1.1 GB of unavoidable HBM traffic vs ~2.5
// GFLOP). All per-graph intermediates (dinv 400KB, xw 6.4MB, agg1 6.4MB,
// xw2 400KB) live in d_ws and fit in the 192MB L2, so scatter atomics and
// gathers are L2-resident. The x@W1 GEMM uses V_WMMA_F32_16X16X4_F32
// (full f32 precision, N=16 == exactly one WMMA tile wide).
// ---------------------------------------------------------------------------

typedef __attribute__((ext_vector_type(2))) float v2f;
typedef __attribute__((ext_vector_type(8))) float v8f;

#define N_GRAPHS 5
#define N_NODES  100000
#define N_EDGES  3200000
#define F_IN     128
#define H1       16

// workspace layout, in floats
#define WS_DINV  0
#define WS_XW    (WS_DINV + N_NODES)        // 100000
#define WS_AGG1  (WS_XW   + N_NODES * H1)   // 1,700,000
#define WS_XW2   (WS_AGG1 + N_NODES * H1)   // 3,300,000
#define WS_SCORE (WS_XW2  + N_NODES)        // 3,400,000  (5 floats)

// ---------------------------------------------------------------------------

__global__ __launch_bounds__(32) void k_zero_scores(float* __restrict__ score) {
    if (threadIdx.x < N_GRAPHS) score[threadIdx.x] = 0.0f;
}

// deg starts at 1.0 (the self-loop added by GCNConv)
__global__ __launch_bounds__(256) void k_init_deg(float* __restrict__ deg) {
    int i = blockIdx.x * 256 + threadIdx.x;
    if (i < N_NODES) deg[i] = 1.0f;
}

// deg[col] += 1 per edge
__global__ __launch_bounds__(256) void k_deg(const long long* __restrict__ ei,
                                             float* __restrict__ deg) {
    int e = blockIdx.x * 256 + threadIdx.x;
    if (e < N_EDGES) {
        int c = (int)ei[N_EDGES + e];   // targets are row 1 of [2,E]
        atomicAdd(&deg[c], 1.0f);
    }
}

// deg -> D^{-1/2}, in place
__global__ __launch_bounds__(256) void k_dinv(float* __restrict__ deg) {
    int i = blockIdx.x * 256 + threadIdx.x;
    if (i < N_NODES) deg[i] = rsqrtf(fmaxf(deg[i], 1e-12f));
}

// xw[N,16] = x[N,128] @ W1[128,16] via V_WMMA_F32_16X16X4_F32.
// One wave per 16-row tile; 8 waves (256 thr) per block. 100000 = 16*6250,
// so every tile is full. Tile guard is wave-uniform => EXEC all-1s for WMMA.
//
// A (16x4 f32) per-lane layout (ISA 7.12.2): lanes 0-15 rows M=0..15 with
// VGPR0=K0,VGPR1=K1; lanes 16-31 same rows with K2,K3  -> an aligned float2
// load x[row][k0+2*half .. +1] per lane per k-chunk.
// B (4x16 f32): VGPR v, lanes 0-15 = row K=v (cols 0-15), lanes 16-31 = K=v+2.
// C/D (16x16 f32): VGPR v, lanes 0-15 = row v, lanes 16-31 = row v+8.
__global__ __launch_bounds__(256) void k_gemm_xw1(const float* __restrict__ x,
                                                  const float* __restrict__ W1,
                                                  float* __restrict__ xw) {
    const int lane = threadIdx.x & 31;
    const int wave = threadIdx.x >> 5;
    const int tile = blockIdx.x * 8 + wave;
    if (tile * 16 >= N_NODES) return;   // wave-uniform

    const int half = lane >> 4;   // 0 or 1
    const int l    = lane & 15;
    const int row  = tile * 16 + l;
    const float* __restrict__ xrow = x + (size_t)row * F_IN;

    v8f c = {};
    #pragma unroll
    for (int k0 = 0; k0 < F_IN; k0 += 4) {
        const int kb = k0 + 2 * half;                 // even -> 8B aligned
        v2f a = *(const v2f*)(xrow + kb);             // A: rows striped in lane
        v2f b;
        b.x = W1[(size_t)kb * H1 + l];                // B row K=kb
        b.y = W1[(size_t)(kb + 1) * H1 + l];          // B row K=kb+1
        c = __builtin_amdgcn_wmma_f32_16x16x4_f32(
                /*neg_a=*/false, a, /*neg_b=*/false, b,
                /*c_mod=*/(short)0, c, /*reuse_a=*/false, /*reuse_b=*/false);
    }

    float* __restrict__ out = xw + (size_t)tile * 16 * H1;
    #pragma unroll
    for (int v = 0; v < 8; ++v)
        out[(v + 8 * half) * H1 + l] = c[v];
}

// self-loop term: agg1[i,:] = xw[i,:] * dinv[i]^2
__global__ __launch_bounds__(256) void k_selfloop1(const float* __restrict__ xw,
                                                   const float* __restrict__ dinv,
                                                   float* __restrict__ agg1) {
    int t = blockIdx.x * 256 + threadIdx.x;   // over N*H1
    if (t < N_NODES * H1) {
        float d = dinv[t >> 4];
        agg1[t] = xw[t] * d * d;
    }
}

// layer-1 edge scatter: 16 threads per edge (one per feature).
// agg1 is 6.4MB -> atomics resolve in L2, not HBM.
__global__ __launch_bounds__(256) void k_scatter1(const long long* __restrict__ ei,
                                                  const float* __restrict__ xw,
                                                  const float* __restrict__ dinv,
                                                  float* __restrict__ agg1) {
    int t = blockIdx.x * 256 + threadIdx.x;
    int e = t >> 4, f = t & 15;
    if (e < N_EDGES) {
        int r = (int)ei[e];
        int c = (int)ei[N_EDGES + e];
        float nrm = dinv[r] * dinv[c];
        atomicAdd(&agg1[(size_t)c * H1 + f], xw[(size_t)r * H1 + f] * nrm);
    }
}

// layer-2 node pass: h = relu(agg1 + b1); xw2 = h . W2; also accumulate the
// self-loop contribution xw2[i]*dinv[i]^2 into score (block-reduced).
__global__ __launch_bounds__(256) void k_node2(const float* __restrict__ agg1,
                                               const float* __restrict__ dinv,
                                               const float* __restrict__ b1,
                                               const float* __restrict__ W2,
                                               float* __restrict__ xw2,
                                               float* __restrict__ score) {
    __shared__ float red[256];
    int i = blockIdx.x * 256 + threadIdx.x;
    float s = 0.0f;
    if (i < N_NODES) {
        float acc = 0.0f;
        #pragma unroll
        for (int k = 0; k < H1; ++k) {
            float h = fmaxf(agg1[(size_t)i * H1 + k] + b1[k], 0.0f);
            acc += h * W2[k];
        }
        xw2[i] = acc;
        float d = dinv[i];
        s = acc * d * d;
    }
    red[threadIdx.x] = s;
    __syncthreads();
    for (int off = 128; off > 0; off >>= 1) {
        if (threadIdx.x < off) red[threadIdx.x] += red[threadIdx.x + off];
        __syncthreads();
    }
    if (threadIdx.x == 0) atomicAdd(score, red[0]);
}

// layer-2 "aggregation": mean over nodes of the scatter == plain sum over
// edges of xw2[row]*dinv[row]*dinv[col]. No scatter needed. Block-reduce,
// one atomic per block.
__global__ __launch_bounds__(256) void k_edge2(const long long* __restrict__ ei,
                                               const float* __restrict__ xw2,
                                               const float* __restrict__ dinv,
                                               float* __restrict__ score) {
    __shared__ float red[256];
    int e = blockIdx.x * 256 + threadIdx.x;
    float s = 0.0f;
    if (e < N_EDGES) {
        int r = (int)ei[e];
        int c = (int)ei[N_EDGES + e];
        s = xw2[r] * dinv[r] * dinv[c];
    }
    red[threadIdx.x] = s;
    __syncthreads();
    for (int off = 128; off > 0; off >>= 1) {
        if (threadIdx.x < off) red[threadIdx.x] += red[threadIdx.x + off];
        __syncthreads();
    }
    if (threadIdx.x == 0) atomicAdd(score, red[0]);
}

// scores -> mean (+b2) -> softmax over the 5 graphs
__global__ __launch_bounds__(32) void k_softmax(const float* __restrict__ score,
                                                const float* __restrict__ b2,
                                                float* __restrict__ out) {
    if (threadIdx.x == 0) {
        float v[N_GRAPHS];
        float m = -1e30f;
        for (int g = 0; g < N_GRAPHS; ++g) {
            v[g] = score[g] * (1.0f / (float)N_NODES) + b2[0];
            m = fmaxf(m, v[g]);
        }
        float sum = 0.0f;
        for (int g = 0; g < N_GRAPHS; ++g) { v[g] = __expf(v[g] - m); sum += v[g]; }
        float inv = 1.0f / sum;
        for (int g = 0; g < N_GRAPHS; ++g) out[g] = v[g] * inv;
    }
}

// ---------------------------------------------------------------------------

extern "C" void kernel_launch(void* const* d_in, const int* in_sizes, int n_in,
                              void* d_out, int out_size, void* d_ws, size_t ws_size,
                              hipStream_t stream) {
    const float*     x  = (const float*)d_in[0];
    const long long* ei = (const long long*)d_in[1];   // int64 per reference
    const float*     W1 = (const float*)d_in[2];
    const float*     b1 = (const float*)d_in[3];
    const float*     W2 = (const float*)d_in[4];
    const float*     b2 = (const float*)d_in[5];
    float* out = (float*)d_out;

    float* ws    = (float*)d_ws;
    float* dinv  = ws + WS_DINV;
    float* xw    = ws + WS_XW;
    float* agg1  = ws + WS_AGG1;
    float* xw2   = ws + WS_XW2;
    float* score = ws + WS_SCORE;

    k_zero_scores<<<1, 32, 0, stream>>>(score);

    const int nb_nodes = (N_NODES + 255) / 256;
    const int nb_edges = (N_EDGES + 255) / 256;
    const int nb_nf    = (N_NODES * H1 + 255) / 256;
    const int nb_ef    = (N_EDGES * 16 + 255) / 256;
    const int n_tiles  = N_NODES / 16;                 // 6250, exact
    const int nb_gemm  = (n_tiles + 7) / 8;

    for (int g = 0; g < N_GRAPHS; ++g) {
        const float*     xg  = x  + (size_t)g * N_NODES * F_IN;
        const long long* eig = ei + (size_t)g * 2 * N_EDGES;

        k_init_deg <<<nb_nodes, 256, 0, stream>>>(dinv);
        k_deg      <<<nb_edges, 256, 0, stream>>>(eig, dinv);
        k_dinv     <<<nb_nodes, 256, 0, stream>>>(dinv);
        k_gemm_xw1 <<<nb_gemm,  256, 0, stream>>>(xg, W1, xw);
        k_selfloop1<<<nb_nf,    256, 0, stream>>>(xw, dinv, agg1);
        k_scatter1 <<<nb_ef,    256, 0, stream>>>(eig, xw, dinv, agg1);
        k_node2    <<<nb_nodes, 256, 0, stream>>>(agg1, dinv, b1, W2, xw2, score + g);
        k_edge2    <<<nb_edges, 256, 0, stream>>>(eig, xw2, dinv, score + g);
    }

    k_softmax<<<1, 32, 0, stream>>>(score, b2, out);
}